// binaryTreeLogicNet_39702677684769
// MI455X (gfx1250) — compile-verified
//
#include <hip/hip_runtime.h>
#include <hip/hip_bf16.h>
#include <math.h>

#define EPS 1e-6f
#define LDIM 256
#define BATCH 65536
#define SPAD 257   // 257 % 64 == 1 -> bank-spread rows in LDS

typedef __attribute__((ext_vector_type(2))) float v2f;
typedef __attribute__((ext_vector_type(8))) float v8f;

// ===================== Kernel 1: Sinkhorn (single WGP, 320KB LDS) =====================
__global__ __launch_bounds__(1024) void sinkhorn_kernel(const float* __restrict__ logits,
                                                        float* __restrict__ P) {
    __shared__ float M[LDIM * SPAD];            // 256 * 257 * 4 = 263,168 B  (<320KB WGP LDS)
    const int t = threadIdx.x;
    const int g = t >> 2;                       // row (row pass) / column (col pass)
    const int q = t & 3;                        // quarter of the 256-long vector

    for (int i = t; i < LDIM * LDIM; i += 1024)
        M[(i >> 8) * SPAD + (i & 255)] = logits[i];
    __syncthreads();

    for (int it = 0; it < 10; ++it) {
        // ---- row logsumexp (axis=1) ----
        {
            const int base = g * SPAD + q * 64;
            float mx = -3.4e38f;
            #pragma unroll 8
            for (int i = 0; i < 64; ++i) mx = fmaxf(mx, M[base + i]);
            mx = fmaxf(mx, __shfl_xor(mx, 1));
            mx = fmaxf(mx, __shfl_xor(mx, 2));
            float s = 0.0f;
            #pragma unroll 8
            for (int i = 0; i < 64; ++i) s += expf(M[base + i] - mx);
            s += __shfl_xor(s, 1);
            s += __shfl_xor(s, 2);
            const float lse = mx + logf(s);
            #pragma unroll 8
            for (int i = 0; i < 64; ++i) M[base + i] -= lse;
        }
        __syncthreads();
        // ---- col logsumexp (axis=0) ----
        {
            const int base = (q * 64) * SPAD + g;
            float mx = -3.4e38f;
            #pragma unroll 8
            for (int i = 0; i < 64; ++i) mx = fmaxf(mx, M[base + i * SPAD]);
            mx = fmaxf(mx, __shfl_xor(mx, 1));
            mx = fmaxf(mx, __shfl_xor(mx, 2));
            float s = 0.0f;
            #pragma unroll 8
            for (int i = 0; i < 64; ++i) s += expf(M[base + i * SPAD] - mx);
            s += __shfl_xor(s, 1);
            s += __shfl_xor(s, 2);
            const float lse = mx + logf(s);
            #pragma unroll 8
            for (int i = 0; i < 64; ++i) M[base + i * SPAD] -= lse;
        }
        __syncthreads();
    }

    for (int i = t; i < LDIM * LDIM; i += 1024)
        P[i] = expf(M[(i >> 8) * SPAD + (i & 255)]);
}

// ===================== Kernel 2: leafT = (x @ P)^T via V_WMMA_F32_16X16X4_F32 =====================
// C(16n x 16m) = A(16n x 4k = P^T tile) * B(4k x 16m = x^T tile), accumulated over k=0..255.
// Block: 256 threads = 8 waves; block covers m0..m0+127 (one wave per 16 batch rows),
// each wave produces all 16 n-tiles. P staged fully in LDS (256 KB).
__global__ __launch_bounds__(256) void gemm_kernel(const float* __restrict__ x,
                                                   const float* __restrict__ P,
                                                   float* __restrict__ leafT) {
    __shared__ float Ps[LDIM * LDIM];           // 256 KB
    const int t = threadIdx.x;
    {
        const float4* Pv = (const float4*)P;
        float4* Sv = (float4*)Ps;
        for (int i = t; i < LDIM * LDIM / 4; i += 256) Sv[i] = Pv[i];
    }
    __syncthreads();

    const int wave  = t >> 5;
    const int lane  = t & 31;
    const int mloc  = lane & 15;                // m within tile (B cols) / n within tile (A rows)
    const int khalf = (lane >> 4) * 2;          // 0 or 2 : K pair held by this lane half
    const int m0    = blockIdx.x * 128 + wave * 16;

    v8f acc[16] = {};                           // 16 n-tiles of 16x16 f32 accumulators

    const float* xrow = x + (size_t)(m0 + mloc) * LDIM + khalf;

    for (int k = 0; k < LDIM; k += 4) {
        v2f bfrag;                              // x^T tile: rows K=khalf,khalf+1 ; col m0+mloc
        bfrag.x = xrow[k];
        bfrag.y = xrow[k + 1];
        #pragma unroll
        for (int nt = 0; nt < 16; ++nt) {
            const int n = nt * 16 + mloc;       // A rows = output n
            v2f afrag;                          // P^T tile: row n ; K=khalf,khalf+1
            afrag.x = Ps[(k + khalf)     * LDIM + n];
            afrag.y = Ps[(k + khalf + 1) * LDIM + n];
            acc[nt] = __builtin_amdgcn_wmma_f32_16x16x4_f32(
                false, afrag, false, bfrag, (short)0, acc[nt], false, false);
        }
    }

    // C/D layout: VGPR i holds rows i (lanes 0-15) and i+8 (lanes 16-31), col = lane&15.
    const size_t col = (size_t)m0 + mloc;
    #pragma unroll
    for (int nt = 0; nt < 16; ++nt) {
        const int nbase = nt * 16 + (lane >> 4) * 8;
        #pragma unroll
        for (int i = 0; i < 8; ++i)
            leafT[(size_t)(nbase + i) * BATCH + col] = acc[nt][i];
    }
}

// ===================== Kernel 3: sequential combine scan =====================
__device__ __forceinline__ float r_poly(float a) {
    float d  = 0.5f - a;
    float d2 = d * d;
    float num = 0.25f + 1.65811f * d + 2.15388f * d2 + 8.2844f * d2 * d + 6.16764f * d2 * d2;
    float den = a * (1.0f - a);
    den = (fabsf(den) < EPS) ? EPS : den;
    return num / den;
}

__device__ __forceinline__ float power_r(float x, float y, float ra, float w0, float w1) {
    ra = (fabsf(ra) < EPS) ? EPS : ra;
    return powf(w0 * powf(x, ra) + w1 * powf(y, ra), 1.0f / ra);
}

__device__ __forceinline__ float F_upper(float x, float y, float a, float w0, float w1) {
    const float xy  = x * y;
    const float mn  = fminf(x, y);
    const float lin = w0 * x + w1 * y;
    float out = lin;                                          // a ~ 0.5 / fallback
    // where-chain in reference order (a is wave-uniform -> no divergence)
    if (a > 0.5f && a < 0.75f) {
        const float R = r_poly(0.75f);
        out = (3.0f - 4.0f * a) * lin + (4.0f * a - 2.0f) * power_r(x, y, R, w0, w1);
    }
    if (a >= 0.75f && a < 1.0f) {
        const float ra = r_poly(fminf(fmaxf(a, 0.75f), 1.0f - EPS));
        out = power_r(x, y, ra, w0, w1);
    }
    if (fabsf(a - 1.0f)  < EPS) out = mn;
    if (a > 1.0f  && a < 1.25f) out = 4.0f * ((1.25f - a) * mn + (a - 1.0f) * xy);
    if (fabsf(a - 1.25f) < EPS) out = xy;
    if (a > 1.25f && a < 2.0f) {
        const float p = sqrtf(fmaxf(3.0f / fmaxf(2.0f - a, EPS) - 1.0f, EPS));
        out = powf(xy, p);
    }
    if (fabsf(a - 2.0f)  < EPS)
        out = ((fabsf(x - 1.0f) < EPS) && (fabsf(y - 1.0f) < EPS)) ? 1.0f : 0.0f;
    return out;
}

__device__ __forceinline__ float F_comb(float x, float y, float a, float w0, float w1) {
    if (isnan(x)) x = EPS;
    x = fminf(fmaxf(x, EPS), 1.0f - EPS);
    if (isnan(y)) y = EPS;
    y = fminf(fmaxf(y, EPS), 1.0f - EPS);
    if (a < 0.5f - EPS) {                                     // rule 8, single-level recursion
        const float af = fminf(fmaxf(1.0f - a, -1.0f + EPS), 2.0f - EPS);
        return 1.0f - F_upper(1.0f - x, 1.0f - y, af, w0, w1);
    }
    return F_upper(x, y, a, w0, w1);
}

__device__ __forceinline__ float combine_op(float l, float r, float a, float w0, float w1) {
    const float nres = F_comb(l, r, a, w0, w1);
    return isnan(nres) ? a : nres;
}

__global__ __launch_bounds__(256) void chain_kernel(const float* __restrict__ leafT,
                                                    const float* __restrict__ weights,
                                                    const float* __restrict__ biases,
                                                    float* __restrict__ out) {
    __shared__ float sA[255], sW0[255], sW1[255];
    const int t = threadIdx.x;
    if (t < 255) {
        const float w0 = weights[2 * t], w1 = weights[2 * t + 1];
        const float m  = fmaxf(w0, w1);
        const float e0 = expf(w0 - m), e1 = expf(w1 - m);
        const float inv = 1.0f / (e0 + e1);
        sW0[t] = e0 * inv;
        sW1[t] = e1 * inv;
        float a = 3.0f / (1.0f + expf(-biases[t])) - 1.0f;    // sigmoid(b)*3 - 1
        sA[t] = fminf(fmaxf(a, -1.0f + EPS), 2.0f - EPS);
    }
    __syncthreads();

    const int b = blockIdx.x * 256 + t;
    float carry = combine_op(leafT[b], leafT[(size_t)BATCH + b], sA[0], sW0[0], sW1[0]);
    for (int l = 2; l < LDIM; ++l) {
        const float leaf = leafT[(size_t)l * BATCH + b];
        carry = combine_op(carry, leaf, sA[l - 1], sW0[l - 1], sW1[l - 1]);
    }
    out[b] = carry;
}

// ===================== Launch =====================
extern "C" void kernel_launch(void* const* d_in, const int* in_sizes, int n_in,
                              void* d_out, int out_size, void* d_ws, size_t ws_size,
                              hipStream_t stream) {
    const float* x       = (const float*)d_in[0];   // (65536, 256)
    const float* logits  = (const float*)d_in[1];   // (256, 256)
    const float* weights = (const float*)d_in[2];   // (255, 2)
    const float* biases  = (const float*)d_in[3];   // (255,)

    float* P     = (float*)d_ws;                    // 256*256 floats
    float* leafT = P + LDIM * LDIM;                 // 256*65536 floats (64 MB)

    sinkhorn_kernel<<<1, 1024, 0, stream>>>(logits, P);
    gemm_kernel<<<BATCH / 128, 256, 0, stream>>>(x, P, leafT);
    chain_kernel<<<BATCH / 256, 256, 0, stream>>>(leafT, weights, biases, (float*)d_out);
}